// GCN_35802847380154
// MI455X (gfx1250) — compile-verified
//
#include <hip/hip_runtime.h>

// GraphSAGE + gated-skip GCN for MI455X (gfx1250, wave32).
// GEMMs: V_WMMA_F32_16X16X4_F32 (full fp32). A-tiles staged into LDS via
// GLOBAL_LOAD_ASYNC_TO_LDS_B128 (ASYNCcnt) when available.

#define NNODES 50000
#define NH 128

typedef __attribute__((ext_vector_type(2))) float v2f;
typedef __attribute__((ext_vector_type(8))) float v8f;
typedef int vint4 __attribute__((vector_size(16)));
typedef __attribute__((address_space(1))) vint4 gvint4;
typedef __attribute__((address_space(3))) vint4 lvint4;

enum { EPI_NONE = 0, EPI_GSC = 1, EPI_GSC_RELU = 2 };

#if defined(__has_builtin)
#if __has_builtin(__builtin_amdgcn_global_load_async_to_lds_b128)
#define HAVE_ASYNC_LDS 1
#endif
#if __has_builtin(__builtin_amdgcn_s_wait_asynccnt)
#define HAVE_WAIT_ASYNC 1
#endif
#endif

__device__ __forceinline__ void wait_asynccnt0() {
#if defined(HAVE_WAIT_ASYNC)
  __builtin_amdgcn_s_wait_asynccnt(0);
#else
  asm volatile("s_wait_asynccnt 0x0" ::: "memory");
#endif
}

__global__ void zero_f32(float* __restrict__ p, long n) {
  long i = (long)blockIdx.x * blockDim.x + threadIdx.x;
  if (i < n) p[i] = 0.0f;
}

__global__ void deg_count(const int* __restrict__ row, float* __restrict__ deg, int e) {
  int i = blockIdx.x * 256 + threadIdx.x;
  if (i < e) atomicAdd(&deg[row[i]], 1.0f);
}

__global__ void inv_deg_k(float* __restrict__ deg, int n) {
  int i = blockIdx.x * 256 + threadIdx.x;
  if (i < n) deg[i] = 1.0f / fmaxf(deg[i], 1.0f);
}

// One wave32 per edge: float4 gather of src[col], 4 atomic f32 adds into dst[row].
template <int D>
__global__ __launch_bounds__(256) void scatter_add(
    const float* __restrict__ src, const int* __restrict__ row,
    const int* __restrict__ col, float* __restrict__ dst, int e) {
  int wv = threadIdx.x >> 5;
  int ln = threadIdx.x & 31;
  int eid = blockIdx.x * 8 + wv;
  if (eid >= e) return;
  const float* sp = src + (long)col[eid] * D;
  float* dp = dst + (long)row[eid] * D;
#pragma unroll
  for (int f = ln * 4; f < D; f += 128) {
    float4 v = *(const float4*)(sp + f);
    atomicAdd(dp + f + 0, v.x);
    atomicAdd(dp + f + 1, v.y);
    atomicAdd(dp + f + 2, v.z);
    atomicAdd(dp + f + 3, v.w);
  }
}

// Stage a 16 x K row-tile (16B chunks) into LDS with padded stride KP.
template <int K, int KP>
__device__ __forceinline__ void stage_tile(const float* __restrict__ g, int row0,
                                           float* __restrict__ l) {
  constexpr int CPR = (K > 0) ? (K / 4) : 1;  // 16B chunks per row
#if defined(HAVE_ASYNC_LDS)
  for (int i = threadIdx.x; i < 16 * CPR; i += 256) {
    int r = i / CPR, c = (i % CPR) * 4;
    __builtin_amdgcn_global_load_async_to_lds_b128(
        (gvint4*)(g + (long)(row0 + r) * K + c),
        (lvint4*)(l + r * KP + c), 0, 0);
  }
#else
  for (int i = threadIdx.x; i < 16 * CPR; i += 256) {
    int r = i / CPR, c = (i % CPR) * 4;
    float4 v = *(const float4*)(g + (long)(row0 + r) * K + c);
    *(float4*)(l + r * KP + c) = v;
  }
#endif
}

// out[N,128] = A1@W1 (+ (A2 * rowscale)@W2) (+ bias)   [EPI_NONE]
// or GSC epilogue: z = sigmoid(g_in + acc + b_co); r = z*o + (1-z)*xsc; (relu)
// 256 threads = 8 waves; block covers 16 rows x 128 cols; wave w -> cols [16w,16w+16).
template <int K1, int K2, int EPI>
__global__ __launch_bounds__(256) void gemm_wmma(
    const float* __restrict__ A1, const float* __restrict__ W1,
    const float* __restrict__ A2, const float* __restrict__ W2,
    const float* __restrict__ rowscale, const float* __restrict__ bias,
    const float* __restrict__ g_in, const float* __restrict__ xsc,
    float* __restrict__ out) {
  constexpr int KP1 = K1 + 4;  // +4 floats: kills bank conflicts, keeps 16B row align
  constexpr int KP2 = (K2 > 0 ? K2 : 4) + 4;
  __shared__ float lds[16 * KP1 + ((K2 > 0) ? 16 * KP2 : 1)];
  float* ldsA1 = lds;
  float* ldsA2 = lds + 16 * KP1;

  const int row0 = blockIdx.x * 16;

  stage_tile<K1, KP1>(A1, row0, ldsA1);
  if constexpr (K2 > 0) stage_tile<K2, KP2>(A2, row0, ldsA2);
#if defined(HAVE_ASYNC_LDS)
  wait_asynccnt0();
#endif
  __syncthreads();

  const int lane = threadIdx.x & 31;
  const int col0 = (threadIdx.x >> 5) * 16;
  const int m = lane & 15;
  const int ks = (lane < 16) ? 0 : 2;  // lanes 0-15: K={0,1}; lanes 16-31: K={2,3}

  v8f acc = {0.f, 0.f, 0.f, 0.f, 0.f, 0.f, 0.f, 0.f};

#pragma unroll 8
  for (int k = 0; k < K1; k += 4) {
    v2f a = {ldsA1[m * KP1 + k + ks], ldsA1[m * KP1 + k + ks + 1]};
    v2f b = {W1[(k + ks) * NH + col0 + m], W1[(k + ks + 1) * NH + col0 + m]};
    acc = __builtin_amdgcn_wmma_f32_16x16x4_f32(false, a, false, b, (short)0, acc,
                                                false, false);
  }
  if constexpr (K2 > 0) {
    const float sc = rowscale[row0 + m];  // A-fragment row == m for both lane halves
#pragma unroll 8
    for (int k = 0; k < K2; k += 4) {
      v2f a = {ldsA2[m * KP2 + k + ks] * sc, ldsA2[m * KP2 + k + ks + 1] * sc};
      v2f b = {W2[(k + ks) * NH + col0 + m], W2[(k + ks + 1) * NH + col0 + m]};
      acc = __builtin_amdgcn_wmma_f32_16x16x4_f32(false, a, false, b, (short)0, acc,
                                                  false, false);
    }
  }

  // C/D layout: VGPR j -> row M=j (lanes 0-15, N=lane) and M=j+8 (lanes 16-31).
  const int halo = (lane >> 4) << 3;
  const int ccol = col0 + m;
  const float bv = bias ? bias[ccol] : 0.0f;  // hoisted: invariant across j
#pragma unroll
  for (int j = 0; j < 8; ++j) {
    int rt = j + halo;
    long idx = (long)(row0 + rt) * NH + ccol;
    float v = acc[j];
    if (EPI == EPI_NONE) {
      out[idx] = v + bv;
    } else {
      float u = g_in[idx] + v + bv;  // bv == b_co[ccol]
      float z = 1.0f / (1.0f + __expf(-u));
      float o = ldsA1[rt * KP1 + ccol];  // K1==128 for GSC epilogues, ccol<128
      float r = z * o + (1.0f - z) * xsc[idx];
      if (EPI == EPI_GSC_RELU) r = fmaxf(r, 0.0f);
      out[idx] = r;
    }
  }
}

extern "C" void kernel_launch(void* const* d_in, const int* in_sizes, int n_in,
                              void* d_out, int out_size, void* d_ws, size_t ws_size,
                              hipStream_t stream) {
  const float* x   = (const float*)d_in[0];
  const int*   row = (const int*)d_in[1];
  const int*   col = (const int*)d_in[2];
  const float* Wsc = (const float*)d_in[3];
  const float* Wci = (const float*)d_in[4];
  const float* bci = (const float*)d_in[5];
  const float* Wco = (const float*)d_in[6];
  const float* bco = (const float*)d_in[7];
  const float* W0  = (const float*)d_in[8];
  const float* b0  = (const float*)d_in[9];
  const float* W1  = (const float*)d_in[10];
  const float* b1  = (const float*)d_in[11];
  const float* W2  = (const float*)d_in[12];
  const float* b2  = (const float*)d_in[13];
  const float* Wf  = (const float*)d_in[14];
  const float* bf  = (const float*)d_in[15];
  float* outp = (float*)d_out;

  const int Nn = NNODES;
  const int E_ = in_sizes[1];

  float* ws  = (float*)d_ws;
  float* inv = ws;                    // N        (deg -> inv_deg, in place)
  float* xsc = inv + Nn;              // N*128    (x @ W_sc, hoisted)
  float* gin = xsc + (long)Nn * 128;  // N*128    (xsc @ W_ci + b_ci, hoisted)
  float* agg = gin + (long)Nn * 128;  // N*256    (segment-sum buffer)
  float* t0  = agg + (long)Nn * 256;  // N*128    (sage outputs o1/o4/o7)
  float* t1  = t0 + (long)Nn * 128;   // N*128    (gsc outputs o3/o6/o8)

  dim3 blk(256);
  const int gN16 = Nn / 16;  // 3125, exact
  const long a256 = (long)Nn * 256, a128 = (long)Nn * 128;

  // degrees -> inv_deg
  zero_f32<<<(Nn + 255) / 256, blk, 0, stream>>>(inv, Nn);
  deg_count<<<(E_ + 255) / 256, blk, 0, stream>>>(row, inv, E_);
  inv_deg_k<<<(Nn + 255) / 256, blk, 0, stream>>>(inv, Nn);

  // hoisted skip-path invariants
  gemm_wmma<256, 0, EPI_NONE><<<gN16, blk, 0, stream>>>(
      x, Wsc, nullptr, nullptr, nullptr, nullptr, nullptr, nullptr, xsc);
  gemm_wmma<128, 0, EPI_NONE><<<gN16, blk, 0, stream>>>(
      xsc, Wci, nullptr, nullptr, nullptr, bci, nullptr, nullptr, gin);

  // layer 0: agg = segsum(x); o1 = x@W0_top + (agg*inv)@W0_bot + b0; o3 = relu(gsc(o1))
  zero_f32<<<(int)((a256 + 255) / 256), blk, 0, stream>>>(agg, a256);
  scatter_add<256><<<(E_ + 7) / 8, blk, 0, stream>>>(x, row, col, agg, E_);
  gemm_wmma<256, 256, EPI_NONE><<<gN16, blk, 0, stream>>>(
      x, W0, agg, W0 + 256 * NH, inv, b0, nullptr, nullptr, t0);
  gemm_wmma<128, 0, EPI_GSC_RELU><<<gN16, blk, 0, stream>>>(
      t0, Wco, nullptr, nullptr, nullptr, bco, gin, xsc, t1);

  // layer 1
  zero_f32<<<(int)((a128 + 255) / 256), blk, 0, stream>>>(agg, a128);
  scatter_add<128><<<(E_ + 7) / 8, blk, 0, stream>>>(t1, row, col, agg, E_);
  gemm_wmma<128, 128, EPI_NONE><<<gN16, blk, 0, stream>>>(
      t1, W1, agg, W1 + 128 * NH, inv, b1, nullptr, nullptr, t0);
  gemm_wmma<128, 0, EPI_GSC_RELU><<<gN16, blk, 0, stream>>>(
      t0, Wco, nullptr, nullptr, nullptr, bco, gin, xsc, t1);

  // layer 2 (no relu on final gsc)
  zero_f32<<<(int)((a128 + 255) / 256), blk, 0, stream>>>(agg, a128);
  scatter_add<128><<<(E_ + 7) / 8, blk, 0, stream>>>(t1, row, col, agg, E_);
  gemm_wmma<128, 128, EPI_NONE><<<gN16, blk, 0, stream>>>(
      t1, W2, agg, W2 + 128 * NH, inv, b2, nullptr, nullptr, t0);
  gemm_wmma<128, 0, EPI_GSC><<<gN16, blk, 0, stream>>>(
      t0, Wco, nullptr, nullptr, nullptr, bco, gin, xsc, t1);

  // final fc -> d_out
  gemm_wmma<128, 0, EPI_NONE><<<gN16, blk, 0, stream>>>(
      t1, Wf, nullptr, nullptr, nullptr, bf, nullptr, nullptr, outp);
}